// QuantumGenerator_33663953666828
// MI455X (gfx1250) — compile-verified
//
#include <hip/hip_runtime.h>
#include <stdint.h>

#define NQ 14
#define NSTATE (1 << NQ)      // 16384 amplitudes
#define NGEN 8
#define QBATCH 512
#define TPB 256               // 8 waves (wave32)
#define EPT (NSTATE / TPB)    // 64 elements per thread

typedef float v2f __attribute__((ext_vector_type(2)));
typedef float v8f __attribute__((ext_vector_type(8)));

// LDS bank swizzle: fold conflict-causing high address bits {6,7,9,11,13}
// into bank bits {2..5}. GF(2)-linear -> XOR-decomposes; folds into literal
// constants in unrolled loops. Makes gate0/1/2 loads+stores bank-conflict-free
// (verified per access pattern); preserves 4-float contiguity for b128.
__host__ __device__ constexpr int SIG(int j) {
  return (((j >> 6) & 1) * 0x14) ^ (((j >> 7) & 1) * 0x28) ^
         (((j >> 9) & 1) * 0x10) ^ (((j >> 11) & 1) * 0x20) ^
         (((j >> 13) & 1) * 0x30);
}
__device__ __forceinline__ int swz(int j) { return j ^ SIG(j); }

// Opacity barrier: forces per-iteration recomputation of derived addresses so
// LICM cannot hoist ~200 loop-invariant XOR'd addresses out of the rep loop
// (round-4 failure mode: >256 VGPRs + s_set_vgpr_msb churn).
__device__ __forceinline__ int opaque(int v) {
  asm volatile("" : "+v"(v));
  return v;
}

// Composed CNOT-ring permutation: dest j reads from source cnot_perm(j).
// Each CNOT is j ^= bit_c(j)<<pos_t (GF(2)-linear), composed innermost-first.
// Qubit q sits at index bit (13-q).
__device__ __forceinline__ uint32_t cnot_perm(uint32_t j) {
#pragma unroll
  for (int i = NQ - 1; i >= 0; --i) {
    const int c = i;
    const int t = (i + 1) % NQ;
    j ^= ((j >> (NQ - 1 - c)) & 1u) << (NQ - 1 - t);
  }
  return j;
}

// Apply a 16x16 gate G (row-major in LDS) on a 4-qubit group whose lowest
// qubit has stride 2^LS. D = G * S_tile as four chained V_WMMA_F32_16X16X4_F32
// (exact f32). A 16x4 layout: lanes 0-15 rows; VGPR0 = K{0,2} per lane half,
// VGPR1 = K{1,3}. B 4x16 mirrors A; C/D: VGPR v = rows v / v+8 per lane half.
// In-place safe: each 16-fiber tile is read and written only by its own wave.
template <int LS>
__device__ __forceinline__ void apply_group_wmma(float* sS, const float* G) {
  const int lane = threadIdx.x & 31;
  const int wave = threadIdx.x >> 5;
  const int row  = lane & 15;
  const int hf   = lane >> 4;
  const int kofs = hf * 2;
  const int sKof = hf ? ((2 << LS) ^ SIG(2 << LS)) : 0;
  const int sMof = hf ? ((8 << LS) ^ SIG(8 << LS)) : 0;
  constexpr int s1 = (1 << LS) ^ SIG(1 << LS);
  v2f a[4];
#pragma unroll
  for (int k = 0; k < 4; ++k) {
    a[k].x = G[row * 16 + 4 * k + kofs + 0];
    a[k].y = G[row * 16 + 4 * k + kofs + 1];
  }
#pragma unroll
  for (int t8 = 0; t8 < 8; ++t8) {           // 64 tiles, 8 per wave
    const int f  = (wave * 8 + t8) * 16 + row;
    const int sb =
        opaque(swz(((f >> LS) << (LS + 4)) + (f & ((1 << LS) - 1))));
    v8f acc = {0.f, 0.f, 0.f, 0.f, 0.f, 0.f, 0.f, 0.f};
#pragma unroll
    for (int k = 0; k < 4; ++k) {
      const int a0 = sb ^ (((4 * k) << LS) ^ SIG((4 * k) << LS)) ^ sKof;
      v2f bf;
      bf.x = sS[a0];
      bf.y = sS[a0 ^ s1];
      acc = __builtin_amdgcn_wmma_f32_16x16x4_f32(false, a[k], false, bf,
                                                  (short)0, acc, false, false);
    }
#pragma unroll
    for (int v = 0; v < 8; ++v)
      sS[sb ^ ((v << LS) ^ SIG(v << LS)) ^ sMof] = acc[v];
  }
}

__global__ __launch_bounds__(TPB) void qgen_circuit_kernel(
    const float* __restrict__ x, const float* __restrict__ W,
    const float* __restrict__ bvec, float* __restrict__ out) {
  __shared__ float    sS[NSTATE];     // 64 KB statevector (bank-swizzled)
  __shared__ float    sG[2][3][256];  // 16x16 group gates, both angle sets
  __shared__ float    sG3[2][16];     // 4x4 gate for qubits 12,13
  __shared__ float    sTrig[4][NQ];   // c(th/2), s(th/2), c(th), s(th)
  __shared__ float    sZ[NQ];
  __shared__ uint32_t sFhi[EPT];      // swz(F(e<<8)) gather table
  __shared__ float    sAhi[EPT];      // product-state amp, qubits 0..5

  const int tid = threadIdx.x;
  const int wg  = blockIdx.x;         // wg = b*NGEN + g
  const int b   = wg >> 3;
  const int g   = wg & (NGEN - 1);

  // theta[p] = sum_q x[b,q] * W[g,q,p] + bias[g,p]; trig for both angle sets
  if (tid < NQ) {
    float th = bvec[g * NQ + tid];
#pragma unroll
    for (int q = 0; q < NQ; ++q)
      th += x[b * NQ + q] * W[(g * NQ + q) * NQ + tid];
    float s1v, c1v, s2v, c2v;
    sincosf(0.5f * th, &s1v, &c1v);   // RY(theta) entries
    sincosf(th, &s2v, &c2v);          // RY(2*theta) entries
    sTrig[0][tid] = c1v; sTrig[1][tid] = s1v;
    sTrig[2][tid] = c2v; sTrig[3][tid] = s2v;
    sZ[tid] = 0.f;
  }
  if (tid < EPT)
    sFhi[tid] = (uint32_t)swz((int)cnot_perm((uint32_t)tid << 8));
  __syncthreads();

  // Initial state = RY(theta)|0..0> = separable product (replaces 1st layer).
  if (tid < EPT) {                    // e covers index bits 13..8 => qubits 0..5
    float p = 1.f;
#pragma unroll
    for (int q = 0; q < 6; ++q) {
      const int bit = (tid >> (5 - q)) & 1;
      p *= bit ? sTrig[1][q] : sTrig[0][q];
    }
    sAhi[tid] = p;
  }
  float alo = 1.f;                    // tid covers bits 7..0 => qubits 6..13
#pragma unroll
  for (int q = 6; q < NQ; ++q) {
    const int bit = (tid >> (NQ - 1 - q)) & 1;
    alo *= bit ? sTrig[1][q] : sTrig[0][q];
  }

  // Grouped gates: G = RY(q)⊗RY(q+1)⊗RY(q+2)⊗RY(q+3) (16x16, row-major)
  {
    const int r = tid >> 4, c = tid & 15;
#pragma unroll
    for (int sel = 0; sel < 2; ++sel) {
#pragma unroll
      for (int grp = 0; grp < 3; ++grp) {
        float p = 1.f;
#pragma unroll
        for (int k = 0; k < 4; ++k) {
          const int q = grp * 4 + k;
          const float C = sTrig[sel * 2 + 0][q];
          const float S = sTrig[sel * 2 + 1][q];
          const int rb = (r >> (3 - k)) & 1;
          const int cb = (c >> (3 - k)) & 1;
          p *= rb ? (cb ? C : S) : (cb ? -S : C);  // [[C,-S],[S,C]]
        }
        sG[sel][grp][tid] = p;
      }
    }
    if (tid < 32) {
      const int sel = tid >> 4, idx = tid & 15;
      const int r2 = idx >> 2, c2 = idx & 3;
      float p = 1.f;
#pragma unroll
      for (int k = 0; k < 2; ++k) {
        const int q = 12 + k;
        const float C = sTrig[sel * 2 + 0][q];
        const float S = sTrig[sel * 2 + 1][q];
        const int rb = (r2 >> (1 - k)) & 1;
        const int cb = (c2 >> (1 - k)) & 1;
        p *= rb ? (cb ? C : S) : (cb ? -S : C);
      }
      sG3[sel][idx] = p;
    }
  }
  __syncthreads();

  const int sTidSw = swz(tid);        // swizzle sourced from tid bits 6,7
#pragma unroll
  for (int e = 0; e < EPT; ++e)
    sS[sTidSw ^ ((e << 8) ^ SIG(e << 8))] = sAhi[e] * alo;
  __syncthreads();

  const int swzFtid = swz((int)cnot_perm((uint32_t)tid));

  // Fused circuit: [CNOT, RY(2θ)] x3 then [CNOT, RY(θ)]  (RY(θ)·RY(θ)=RY(2θ);
  // first RY(θ) absorbed into the product-state init above).
#pragma unroll 1
  for (int rep = 0; rep < 4; ++rep) {
    // CNOT ring = fixed GF(2)-linear permutation gather (register-staged)
    {
      const int pfb = opaque(swzFtid);
      const int psb = opaque(sTidSw);
      float val[EPT];
#pragma unroll
      for (int e = 0; e < EPT; ++e)
        val[e] = sS[pfb ^ (int)sFhi[e]];
      __syncthreads();
#pragma unroll
      for (int e = 0; e < EPT; ++e)
        sS[psb ^ ((e << 8) ^ SIG(e << 8))] = val[e];
      __syncthreads();
    }

    const int sel = (rep < 3) ? 1 : 0;
    apply_group_wmma<10>(sS, &sG[sel][0][0]);  // qubits 0..3
    __syncthreads();
    apply_group_wmma<6>(sS, &sG[sel][1][0]);   // qubits 4..7
    __syncthreads();
    apply_group_wmma<2>(sS, &sG[sel][2][0]);   // qubits 8..11
    __syncthreads();

    // qubits 12,13: 4x4 gate on contiguous 4-element fibers (VALU, b128)
    {
      const int g3b = opaque(swz(tid << 2));
      float m[16];
#pragma unroll
      for (int i = 0; i < 16; ++i) m[i] = sG3[sel][i];
#pragma unroll
      for (int e = 0; e < EPT / 4; ++e) {
        const int fb = g3b ^ ((e << 10) ^ SIG(e << 10));
        const float v0 = sS[fb + 0], v1 = sS[fb + 1];
        const float v2 = sS[fb + 2], v3 = sS[fb + 3];
        sS[fb + 0] = m[0]  * v0 + m[1]  * v1 + m[2]  * v2 + m[3]  * v3;
        sS[fb + 1] = m[4]  * v0 + m[5]  * v1 + m[6]  * v2 + m[7]  * v3;
        sS[fb + 2] = m[8]  * v0 + m[9]  * v1 + m[10] * v2 + m[11] * v3;
        sS[fb + 3] = m[12] * v0 + m[13] * v1 + m[14] * v2 + m[15] * v3;
      }
    }
    __syncthreads();
  }

  // <Z_i> = sum_j amp(j)^2 * (1 - 2*bit_i(j)),  j = e*256 + tid.
  // Bit-disjoint split: qubits 0..5 -> sign depends only on e (literal per
  // unrolled iteration); qubits 6..13 -> sign depends only on tid (so their
  // contribution is just +-Psum). Halves the accumulation VALU.
  float zs[6];
#pragma unroll
  for (int i = 0; i < 6; ++i) zs[i] = 0.f;
  float psum = 0.f;
#pragma unroll
  for (int e = 0; e < EPT; ++e) {
    const float v = sS[sTidSw ^ ((e << 8) ^ SIG(e << 8))];
    const float p = v * v;
    psum += p;
#pragma unroll
    for (int i = 0; i < 6; ++i)
      zs[i] += ((e >> (5 - i)) & 1) ? -p : p;   // literal sign per e
  }
  float z[NQ];
#pragma unroll
  for (int i = 0; i < 6; ++i) z[i] = zs[i];
#pragma unroll
  for (int i = 6; i < NQ; ++i)
    z[i] = ((tid >> (NQ - 1 - i)) & 1) ? -psum : psum;
#pragma unroll
  for (int i = 0; i < NQ; ++i) {
    float v = z[i];
#pragma unroll
    for (int off = 16; off >= 1; off >>= 1)
      v += __shfl_xor(v, off, 32);
    if ((tid & 31) == 0) atomicAdd(&sZ[i], v);   // ds_add_f32
  }
  __syncthreads();
  if (tid < NQ) out[wg * NQ + tid] = sZ[tid];
}

extern "C" void kernel_launch(void* const* d_in, const int* in_sizes, int n_in,
                              void* d_out, int out_size, void* d_ws,
                              size_t ws_size, hipStream_t stream) {
  (void)in_sizes; (void)n_in; (void)out_size; (void)d_ws; (void)ws_size;
  const float* x    = (const float*)d_in[0];  // [512,14]
  const float* W    = (const float*)d_in[1];  // [8,14,14]
  const float* bvec = (const float*)d_in[2];  // [8,14]
  float* out = (float*)d_out;                 // [512,112]
  qgen_circuit_kernel<<<dim3(QBATCH * NGEN), dim3(TPB), 0, stream>>>(x, W, bvec,
                                                                    out);
}